// BatchNeuralMemory_89781996355759
// MI455X (gfx1250) — compile-verified
//
#include <hip/hip_runtime.h>
#include <math.h>

// BatchNeuralMemory for MI455X (gfx1250): fp32 WMMA GEMMs, 2x4 register-blocked
// macro-tiles per wave, scan fused into weight-grad epilogues.
// B=2, S=1024, D=1024, C=64, I=2048, N=16.

#define B_ 2
#define S_ 1024
#define D_ 1024
#define C_ 64
#define I_ 2048
#define N_ 16
#define EPSF 1e-6f
#define BASE_LR 0.01f

typedef __attribute__((ext_vector_type(2))) float v2f;
typedef __attribute__((ext_vector_type(8))) float v8f;

// ---------------------------------------------------------------------------
// Register-blocked WMMA macro-tile: MT x NT grid of 16x16 fp32 tiles per wave.
// Logical A[m][k]: TA ? mem[k*lda+m] : mem[m*lda+k]
// Logical B[k][n]: TB ? mem[n*ldb+k] : mem[k*ldb+n]
// A frag: lane L holds M=L%16, K = 2*(L/16)+{0,1}. B frag symmetric (N=L%16).
// C/D frag: lane L, reg r -> row = r + 8*(L/16), col = L%16.
// Per K-step: MT+NT fragment loads feed MT*NT v_wmma_f32_16x16x4_f32.
// ---------------------------------------------------------------------------
template <bool TA, bool TB, int MT, int NT>
__device__ __forceinline__ void wmma_mt(const float* __restrict__ A, int lda,
                                        const float* __restrict__ Bm, int ldb,
                                        int m0, int n0, int K, v8f (&acc)[MT][NT]) {
  const int lane = threadIdx.x & 31;
  const int h2 = (lane >> 4) * 2;
  const int l16 = lane & 15;
#pragma unroll
  for (int mt = 0; mt < MT; ++mt)
#pragma unroll
    for (int nt = 0; nt < NT; ++nt)
      acc[mt][nt] = (v8f){0.f, 0.f, 0.f, 0.f, 0.f, 0.f, 0.f, 0.f};

  // prefetch first rows (global_prefetch_b8)
  if (TA) __builtin_prefetch(A + m0 + l16, 0, 1);
  else    __builtin_prefetch(A + (size_t)(m0 + l16) * lda, 0, 1);
  if (TB) __builtin_prefetch(Bm + (size_t)(n0 + l16) * ldb, 0, 1);
  else    __builtin_prefetch(Bm + n0 + l16, 0, 1);

  for (int k = 0; k < K; k += 4) {
    const int ka = k + h2;
    v2f a[MT], b[NT];
#pragma unroll
    for (int mt = 0; mt < MT; ++mt) {
      const int mr = m0 + mt * 16 + l16;
      if (TA) {
        a[mt][0] = A[(size_t)(ka + 0) * lda + mr];
        a[mt][1] = A[(size_t)(ka + 1) * lda + mr];
      } else {
        a[mt][0] = A[(size_t)mr * lda + ka + 0];
        a[mt][1] = A[(size_t)mr * lda + ka + 1];
      }
    }
#pragma unroll
    for (int nt = 0; nt < NT; ++nt) {
      const int nc = n0 + nt * 16 + l16;
      if (TB) {
        b[nt][0] = Bm[(size_t)nc * ldb + ka + 0];
        b[nt][1] = Bm[(size_t)nc * ldb + ka + 1];
      } else {
        b[nt][0] = Bm[(size_t)(ka + 0) * ldb + nc];
        b[nt][1] = Bm[(size_t)(ka + 1) * ldb + nc];
      }
    }
#pragma unroll
    for (int mt = 0; mt < MT; ++mt)
#pragma unroll
      for (int nt = 0; nt < NT; ++nt)
        acc[mt][nt] = __builtin_amdgcn_wmma_f32_16x16x4_f32(
            false, a[mt], false, b[nt], (short)0, acc[mt][nt], false, false);
  }
}

// Visit every element of the macro-tile: f(row, col, value)
template <int MT, int NT, class F>
__device__ __forceinline__ void tile_foreach(int m0, int n0, v8f (&acc)[MT][NT], F&& f) {
  const int lane = threadIdx.x & 31;
  const int l16 = lane & 15;
  const int rb = (lane >> 4) * 8;
#pragma unroll
  for (int mt = 0; mt < MT; ++mt)
#pragma unroll
    for (int nt = 0; nt < NT; ++nt)
#pragma unroll
      for (int r = 0; r < 8; ++r)
        f(m0 + mt * 16 + rb + r, n0 + nt * 16 + l16, acc[mt][nt][r]);
}

#define MT_ 2
#define NT_ 4

__device__ __forceinline__ float sigm(float x) { return 1.0f / (1.0f + expf(-x)); }
__device__ __forceinline__ float siluf(float x) { return x * sigm(x); }

__device__ __forceinline__ float bsum(float v, float* sm) {
  const int t = threadIdx.x;
  sm[t] = v;
  __syncthreads();
  for (int s = blockDim.x >> 1; s > 0; s >>= 1) {
    if (t < s) sm[t] += sm[t + s];
    __syncthreads();
  }
  const float r = sm[0];
  __syncthreads();
  return r;
}

// ---------------------------------------------------------------------------
// Init scan state: M = broadcast(src) per batch, S = 0.
__global__ void bnm_init_pair(float* Mdst, float* Sdst, const float* src, int nper) {
  const int i = blockIdx.x * blockDim.x + threadIdx.x;
  if (i < nper) {
    const float v = src[i];
    for (int b = 0; b < B_; ++b) {
      Mdst[(size_t)b * nper + i] = v;
      Sdst[(size_t)b * nper + i] = 0.0f;
    }
  }
}

// Gates: layout [alpha(32) | theta(32) | eta(32)]
__global__ void bnm_gates(const float* __restrict__ x, const float* __restrict__ aw,
                          const float* __restrict__ tw, const float* __restrict__ ew,
                          float* gates) {
  __shared__ float sm[256];
  const int bn = blockIdx.x;  // chunk offset bn*C*D into flattened x
  const float* xc = x + (size_t)bn * C_ * D_;
  float sa = 0.f, st = 0.f, se = 0.f;
  for (int i = threadIdx.x; i < C_ * D_; i += blockDim.x) {
    const float xv = xc[i];
    sa += xv * aw[i];
    st += xv * tw[i];
    se += xv * ew[i];
  }
  sa = bsum(sa, sm);
  st = bsum(st, sm);
  se = bsum(se, sm);
  if (threadIdx.x == 0) {
    gates[bn] = sigm(sa);
    gates[B_ * N_ + bn] = sigm(st) * BASE_LR;
    gates[2 * B_ * N_ + bn] = sigm(se);
  }
}

// Y[m,:] = silu(X[m,:] @ W^T), M = B*S, N = D, K = D
__global__ void bnm_proj_silu(const float* __restrict__ X, const float* __restrict__ W,
                              float* __restrict__ Y) {
  const int m0 = blockIdx.y * (16 * MT_), n0 = blockIdx.x * (16 * NT_);
  v8f acc[MT_][NT_];
  wmma_mt<false, true, MT_, NT_>(X, D_, W, D_, m0, n0, D_, acc);
  tile_foreach<MT_, NT_>(m0, n0, acc, [&](int row, int col, float v) {
    Y[(size_t)row * D_ + col] = siluf(v);
  });
}

// In-place rms over rows of length D: p = p * rsqrt(mean(p^2)+eps) * gamma
__global__ void bnm_rms_rows(float* __restrict__ buf, const float* __restrict__ gamma) {
  __shared__ float sm[256];
  float* p = buf + (size_t)blockIdx.x * D_;
  float ss = 0.f;
  for (int i = threadIdx.x; i < D_; i += blockDim.x) { const float v = p[i]; ss += v * v; }
  ss = bsum(ss, sm);
  const float inv = rsqrtf(ss / (float)D_ + EPSF);
  for (int i = threadIdx.x; i < D_; i += blockDim.x) p[i] = p[i] * inv * gamma[i];
}

// pre1 = k_chunk @ w1^T ; h = silu(pre1). M=C per batch, N=I, K=D
__global__ void bnm_chunk_pre_h(const float* __restrict__ kbuf, const float* __restrict__ w1,
                                float* __restrict__ pre1, float* __restrict__ hbuf, int n) {
  const int b = blockIdx.z;
  const float* A = kbuf + ((size_t)b * S_ + n * C_) * D_;
  const int m0 = blockIdx.y * (16 * MT_), n0 = blockIdx.x * (16 * NT_);
  v8f acc[MT_][NT_];
  wmma_mt<false, true, MT_, NT_>(A, D_, w1, D_, m0, n0, D_, acc);
  float* pb = pre1 + (size_t)b * C_ * I_;
  float* hb = hbuf + (size_t)b * C_ * I_;
  tile_foreach<MT_, NT_>(m0, n0, acc, [&](int row, int col, float v) {
    pb[(size_t)row * I_ + col] = v;
    hb[(size_t)row * I_ + col] = siluf(v);
  });
}

// y = h @ w2^T. M=C, N=D, K=I
__global__ void bnm_chunk_y(const float* __restrict__ hbuf, const float* __restrict__ w2,
                            float* __restrict__ ybuf) {
  const int b = blockIdx.z;
  const float* A = hbuf + (size_t)b * C_ * I_;
  const int m0 = blockIdx.y * (16 * MT_), n0 = blockIdx.x * (16 * NT_);
  v8f acc[MT_][NT_];
  wmma_mt<false, true, MT_, NT_>(A, I_, w2, I_, m0, n0, I_, acc);
  float* yb = ybuf + (size_t)b * C_ * D_;
  tile_foreach<MT_, NT_>(m0, n0, acc, [&](int row, int col, float v) {
    yb[(size_t)row * D_ + col] = v;
  });
}

// Per-row backward through rms+loss: produces yn, dpred, dy.
__global__ void bnm_chunk_rowback(const float* __restrict__ ybuf, const float* __restrict__ kbuf,
                                  const float* __restrict__ vbuf, const float* __restrict__ lnw,
                                  const float* __restrict__ gates, float* __restrict__ ynb,
                                  float* __restrict__ dpb, float* __restrict__ dyb, int n) {
  __shared__ float sm[256];
  const int b = blockIdx.x >> 6;  // blocks: b*C + c
  const int c = blockIdx.x & (C_ - 1);
  const float* y = ybuf + ((size_t)b * C_ + c) * D_;
  const float* kk = kbuf + ((size_t)b * S_ + n * C_ + c) * D_;
  const float* vv = vbuf + ((size_t)b * S_ + n * C_ + c) * D_;
  float* ynr = ynb + ((size_t)b * C_ + c) * D_;
  float* dpr = dpb + ((size_t)b * C_ + c) * D_;
  float* dyr = dyb + ((size_t)b * C_ + c) * D_;
  const float theta = gates[B_ * N_ + b * N_ + n];
  float ss = 0.f;
  for (int i = threadIdx.x; i < D_; i += blockDim.x) { const float t = y[i]; ss += t * t; }
  ss = bsum(ss, sm);
  const float inv = rsqrtf(ss / (float)D_ + EPSF);
  const float scale = -2.0f * theta / (float)D_;
  float rd = 0.f;
  for (int i = threadIdx.x; i < D_; i += blockDim.x) {
    const float yn = y[i] * inv;
    const float pred = kk[i] + yn * lnw[i];
    const float dp = scale * (vv[i] - pred);
    ynr[i] = yn;
    dpr[i] = dp;
    rd += (dp * lnw[i]) * yn;
  }
  rd = bsum(rd, sm);
  const float rdm = rd / (float)D_;
  for (int i = threadIdx.x; i < D_; i += blockDim.x) {
    const float yn = ynr[i];
    const float dyn = dpr[i] * lnw[i];
    dyr[i] = inv * (dyn - yn * rdm);
  }
}

// ln-weight grad + scan update: gl[d] = sum_c dpred*yn; S=eta*S-gl; M=(1-a)M+S
__global__ void bnm_scan_ln(const float* __restrict__ dpb, const float* __restrict__ ynb,
                            const float* __restrict__ gates, float* __restrict__ nln,
                            float* __restrict__ nsl, int n) {
  const int idx = blockIdx.x * blockDim.x + threadIdx.x;
  if (idx >= B_ * D_) return;
  const int b = idx / D_, d = idx - b * D_;
  float g = 0.f;
  for (int c = 0; c < C_; ++c)
    g += dpb[((size_t)b * C_ + c) * D_ + d] * ynb[((size_t)b * C_ + c) * D_ + d];
  const float alpha = gates[b * N_ + n];
  const float eta = gates[2 * B_ * N_ + b * N_ + n];
  const float s = eta * nsl[idx] - g;
  nsl[idx] = s;
  nln[idx] = (1.0f - alpha) * nln[idx] + s;
}

// dpre1 = (dy @ w2) * silu'(pre1). M=C, N=I, K=D; B natural (w2[k=d][n=i])
__global__ void bnm_chunk_dh(const float* __restrict__ dyb, const float* __restrict__ w2,
                             const float* __restrict__ pre1, float* __restrict__ dp1b) {
  const int b = blockIdx.z;
  const float* A = dyb + (size_t)b * C_ * D_;
  const int m0 = blockIdx.y * (16 * MT_), n0 = blockIdx.x * (16 * NT_);
  v8f acc[MT_][NT_];
  wmma_mt<false, false, MT_, NT_>(A, D_, w2, I_, m0, n0, D_, acc);
  const float* pb = pre1 + (size_t)b * C_ * I_;
  float* db = dp1b + (size_t)b * C_ * I_;
  tile_foreach<MT_, NT_>(m0, n0, acc, [&](int row, int col, float v) {
    const float p = pb[(size_t)row * I_ + col];
    const float sg = sigm(p);
    db[(size_t)row * I_ + col] = v * sg * (1.0f + p * (1.0f - sg));
  });
}

// g1[i,d] = sum_c dpre1[c,i]*k[c,d]; fused scan into nw1/ns1. M=I, N=D, K=C
__global__ void bnm_scan_w1(const float* __restrict__ dp1b, const float* __restrict__ kbuf,
                            const float* __restrict__ gates, float* __restrict__ nw1,
                            float* __restrict__ ns1, int n) {
  const int b = blockIdx.z;
  const float* A = dp1b + (size_t)b * C_ * I_;               // A[i][c] = mem[c*I+i] (TA)
  const float* Bm = kbuf + ((size_t)b * S_ + n * C_) * D_;   // B[c][d]
  const int m0 = blockIdx.y * (16 * MT_), n0 = blockIdx.x * (16 * NT_);
  v8f acc[MT_][NT_];
  wmma_mt<true, false, MT_, NT_>(A, I_, Bm, D_, m0, n0, C_, acc);
  const float alpha = gates[b * N_ + n];
  const float eta = gates[2 * B_ * N_ + b * N_ + n];
  float* M = nw1 + (size_t)b * I_ * D_;
  float* Sp = ns1 + (size_t)b * I_ * D_;
  tile_foreach<MT_, NT_>(m0, n0, acc, [&](int row, int col, float g) {
    const size_t idx = (size_t)row * D_ + col;
    const float s = eta * Sp[idx] - g;
    Sp[idx] = s;
    M[idx] = (1.0f - alpha) * M[idx] + s;
  });
}

// g2[d,i] = sum_c dy[c,d]*h[c,i]; fused scan into nw2/ns2. M=D, N=I, K=C
__global__ void bnm_scan_w2(const float* __restrict__ dyb, const float* __restrict__ hbuf,
                            const float* __restrict__ gates, float* __restrict__ nw2,
                            float* __restrict__ ns2, int n) {
  const int b = blockIdx.z;
  const float* A = dyb + (size_t)b * C_ * D_;   // A[d][c] = mem[c*D+d] (TA)
  const float* Bm = hbuf + (size_t)b * C_ * I_; // B[c][i]
  const int m0 = blockIdx.y * (16 * MT_), n0 = blockIdx.x * (16 * NT_);
  v8f acc[MT_][NT_];
  wmma_mt<true, false, MT_, NT_>(A, D_, Bm, I_, m0, n0, C_, acc);
  const float alpha = gates[b * N_ + n];
  const float eta = gates[2 * B_ * N_ + b * N_ + n];
  float* M = nw2 + (size_t)b * D_ * I_;
  float* Sp = ns2 + (size_t)b * D_ * I_;
  tile_foreach<MT_, NT_>(m0, n0, acc, [&](int row, int col, float g) {
    const size_t idx = (size_t)row * I_ + col;
    const float s = eta * Sp[idx] - g;
    Sp[idx] = s;
    M[idx] = (1.0f - alpha) * M[idx] + s;
  });
}

// Retrieval: hq = silu(q_chunk @ nw1[b]^T). M=C, N=I, K=D
__global__ void bnm_retr_h(const float* __restrict__ qbuf, const float* __restrict__ nw1,
                           float* __restrict__ hbuf, int n) {
  const int b = blockIdx.z;
  const float* A = qbuf + ((size_t)b * S_ + n * C_) * D_;
  const float* W = nw1 + (size_t)b * I_ * D_;
  const int m0 = blockIdx.y * (16 * MT_), n0 = blockIdx.x * (16 * NT_);
  v8f acc[MT_][NT_];
  wmma_mt<false, true, MT_, NT_>(A, D_, W, D_, m0, n0, D_, acc);
  float* hb = hbuf + (size_t)b * C_ * I_;
  tile_foreach<MT_, NT_>(m0, n0, acc, [&](int row, int col, float v) {
    hb[(size_t)row * I_ + col] = siluf(v);
  });
}

// Retrieval: yq = hq @ nw2[b]^T. M=C, N=D, K=I
__global__ void bnm_retr_y(const float* __restrict__ hbuf, const float* __restrict__ nw2,
                           float* __restrict__ ybuf) {
  const int b = blockIdx.z;
  const float* A = hbuf + (size_t)b * C_ * I_;
  const float* W = nw2 + (size_t)b * D_ * I_;
  const int m0 = blockIdx.y * (16 * MT_), n0 = blockIdx.x * (16 * NT_);
  v8f acc[MT_][NT_];
  wmma_mt<false, true, MT_, NT_>(A, I_, W, I_, m0, n0, I_, acc);
  float* yb = ybuf + (size_t)b * C_ * D_;
  tile_foreach<MT_, NT_>(m0, n0, acc, [&](int row, int col, float v) {
    yb[(size_t)row * D_ + col] = v;
  });
}

// out = q + rms(yq, nln)
__global__ void bnm_retr_out(const float* __restrict__ ybuf, const float* __restrict__ qbuf,
                             const float* __restrict__ nln, float* __restrict__ out, int n) {
  __shared__ float sm[256];
  const int b = blockIdx.x >> 6;
  const int c = blockIdx.x & (C_ - 1);
  const float* y = ybuf + ((size_t)b * C_ + c) * D_;
  const float* q = qbuf + ((size_t)b * S_ + n * C_ + c) * D_;
  float* o = out + ((size_t)b * S_ + n * C_ + c) * D_;
  const float* g = nln + (size_t)b * D_;
  float ss = 0.f;
  for (int i = threadIdx.x; i < D_; i += blockDim.x) { const float t = y[i]; ss += t * t; }
  ss = bsum(ss, sm);
  const float inv = rsqrtf(ss / (float)D_ + EPSF);
  for (int i = threadIdx.x; i < D_; i += blockDim.x) o[i] = q[i] + y[i] * inv * g[i];
}

// ---------------------------------------------------------------------------
extern "C" void kernel_launch(void* const* d_in, const int* in_sizes, int n_in,
                              void* d_out, int out_size, void* d_ws, size_t ws_size,
                              hipStream_t stream) {
  (void)in_sizes; (void)n_in; (void)out_size; (void)ws_size;
  const float* x   = (const float*)d_in[0];
  const float* w1  = (const float*)d_in[1];
  const float* w2  = (const float*)d_in[2];
  const float* lnw = (const float*)d_in[3];
  const float* wq  = (const float*)d_in[4];
  const float* wk  = (const float*)d_in[5];
  const float* wv  = (const float*)d_in[6];
  const float* qnw = (const float*)d_in[7];
  const float* knw = (const float*)d_in[8];
  const float* aw  = (const float*)d_in[9];
  const float* tw  = (const float*)d_in[10];
  const float* ew  = (const float*)d_in[11];

  float* out = (float*)d_out;
  float* ws  = (float*)d_ws;

  const size_t SD = (size_t)S_ * D_;
  // workspace layout (~30.4 MB)
  float* kbuf = ws;
  float* vbuf = kbuf + (size_t)B_ * SD;
  float* qbuf = vbuf + (size_t)B_ * SD;
  float* pre1 = qbuf + (size_t)B_ * SD;
  float* hbuf = pre1 + (size_t)B_ * C_ * I_;
  float* ybuf = hbuf + (size_t)B_ * C_ * I_;
  float* ynb  = ybuf + (size_t)B_ * C_ * D_;
  float* dpb  = ynb  + (size_t)B_ * C_ * D_;
  float* dyb  = dpb  + (size_t)B_ * C_ * D_;
  float* dp1b = dyb  + (size_t)B_ * C_ * D_;
  float* gates = dp1b + (size_t)B_ * C_ * I_;

  // output slices: out, nw1, nw2, nln, ns1, ns2, nsl (scan state lives here)
  float* o_out = out;
  float* nw1 = out + (size_t)B_ * SD;
  float* nw2 = nw1 + (size_t)B_ * I_ * D_;
  float* nln = nw2 + (size_t)B_ * D_ * I_;
  float* ns1 = nln + (size_t)B_ * D_;
  float* ns2 = ns1 + (size_t)B_ * I_ * D_;
  float* nsl = ns2 + (size_t)B_ * D_ * I_;

  const int TM = 16 * MT_;  // 32
  const int TN = 16 * NT_;  // 64

  // 1) init scan state
  bnm_init_pair<<<(I_ * D_ + 255) / 256, 256, 0, stream>>>(nw1, ns1, w1, I_ * D_);
  bnm_init_pair<<<(D_ * I_ + 255) / 256, 256, 0, stream>>>(nw2, ns2, w2, D_ * I_);
  bnm_init_pair<<<(D_ + 255) / 256, 256, 0, stream>>>(nln, nsl, lnw, D_);

  // 2) gates
  bnm_gates<<<B_ * N_, 256, 0, stream>>>(x, aw, tw, ew, gates);

  // 3) projections
  dim3 gp(D_ / TN, (B_ * S_) / TM);
  bnm_proj_silu<<<gp, 32, 0, stream>>>(x, wk, kbuf);
  bnm_rms_rows<<<B_ * S_, 256, 0, stream>>>(kbuf, knw);
  bnm_proj_silu<<<gp, 32, 0, stream>>>(x, wv, vbuf);
  bnm_proj_silu<<<gp, 32, 0, stream>>>(x, wq, qbuf);
  bnm_rms_rows<<<B_ * S_, 256, 0, stream>>>(qbuf, qnw);

  // 4) per-chunk grads (all at initial params) + fused linear scan
  for (int n = 0; n < N_; ++n) {
    bnm_chunk_pre_h<<<dim3(I_ / TN, C_ / TM, B_), 32, 0, stream>>>(kbuf, w1, pre1, hbuf, n);
    bnm_chunk_y<<<dim3(D_ / TN, C_ / TM, B_), 32, 0, stream>>>(hbuf, w2, ybuf);
    bnm_chunk_rowback<<<B_ * C_, 256, 0, stream>>>(ybuf, kbuf, vbuf, lnw, gates, ynb, dpb, dyb, n);
    bnm_scan_ln<<<(B_ * D_ + 255) / 256, 256, 0, stream>>>(dpb, ynb, gates, nln, nsl, n);
    bnm_chunk_dh<<<dim3(I_ / TN, C_ / TM, B_), 32, 0, stream>>>(dyb, w2, pre1, dp1b);
    bnm_scan_w1<<<dim3(D_ / TN, I_ / TM, B_), 32, 0, stream>>>(dp1b, kbuf, gates, nw1, ns1, n);
    bnm_scan_w2<<<dim3(I_ / TN, D_ / TM, B_), 32, 0, stream>>>(dyb, hbuf, gates, nw2, ns2, n);
  }

  // 5) retrieval with updated memory, chunked to reuse scratch
  for (int n = 0; n < N_; ++n) {
    bnm_retr_h<<<dim3(I_ / TN, C_ / TM, B_), 32, 0, stream>>>(qbuf, nw1, hbuf, n);
    bnm_retr_y<<<dim3(D_ / TN, C_ / TM, B_), 32, 0, stream>>>(hbuf, nw2, ybuf);
    bnm_retr_out<<<B_ * C_, 256, 0, stream>>>(ybuf, qbuf, nln, o_out, n);
  }
}